// GCN_11501922419253
// MI455X (gfx1250) — compile-verified
//
#include <hip/hip_runtime.h>
#include <hip/hip_bf16.h>

typedef __attribute__((ext_vector_type(2))) float v2f;
typedef __attribute__((ext_vector_type(8))) float v8f;

#define NUM_GRAPHS 512

// ---------------- zero fill ----------------
__global__ void k_zero(float* __restrict__ p, int n) {
  int i = blockIdx.x * blockDim.x + threadIdx.x;
  int stride = gridDim.x * blockDim.x;
  for (; i < n; i += stride) p[i] = 0.0f;
}

// ---------------- degree on dst (float atomic) ----------------
__global__ void k_deg(const int* __restrict__ dst, float* __restrict__ degf, int E) {
  int e = blockIdx.x * blockDim.x + threadIdx.x;
  if (e >= E) return;
  __builtin_prefetch(dst + e + 8192, 0, 0);
  atomicAdd(&degf[dst[e]], 1.0f);
}

// ---------------- dis = rsqrt(deg+1); xs = x*dis ----------------
__global__ void k_norm(const float* __restrict__ x, const float* __restrict__ degf,
                       float* __restrict__ dis, float* __restrict__ xs, int N) {
  int i = blockIdx.x * blockDim.x + threadIdx.x;
  if (i >= N) return;
  float d = rsqrtf(degf[i] + 1.0f);
  dis[i] = d;
  xs[i] = x[i] * d;
}

// ---------------- layer-1 scalar scatter: s1[dst] += xs[src]*dis[dst] ----------------
__global__ void k_edge1(const int* __restrict__ src, const int* __restrict__ dst,
                        const float* __restrict__ xs, const float* __restrict__ dis,
                        float* __restrict__ s1, int E) {
  int e = blockIdx.x * blockDim.x + threadIdx.x;
  if (e >= E) return;
  __builtin_prefetch(src + e + 8192, 0, 0);
  __builtin_prefetch(dst + e + 8192, 0, 0);
  int s = src[e], d = dst[e];
  atomicAdd(&s1[d], xs[s] * dis[d]);
}

// ---------------- fused: h1 = relu(S*W1 + b1); zz = (h1 @ W2) * dis  via V_WMMA_F32_16X16X4_F32
// One wave computes 16 nodes. A (16x4): lane 0-15 hold M=lane with K={0,1} in {a.x,a.y};
// lanes 16-31 hold M=lane-16 with K={2,3}. B (4x16) mirrored: VGPR0 rows K=0|K=2, VGPR1 K=1|K=3.
// D: VGPR r, lanes 0-15 = (M=r, N=lane); lanes 16-31 = (M=r+8, N=lane-16).
__global__ void k_l1_wmma(const float* __restrict__ x, const float* __restrict__ s1,
                          const float* __restrict__ dis,
                          const float* __restrict__ W1, const float* __restrict__ b1,
                          const float* __restrict__ W2,
                          float* __restrict__ zz, int N) {
  int lane = threadIdx.x & 31;
  int wave = blockIdx.x * (blockDim.x >> 5) + (threadIdx.x >> 5);
  int base = wave * 16;
  int m    = lane & 15;
  int hi   = lane >> 4;
  int k0   = hi ? 2 : 0;

  int node = base + m;
  if (node >= N) node = N - 1;          // clamp loads; EXEC must stay all-ones for WMMA

  float dn = dis[node];
  float S  = s1[node] + x[node] * dn * dn;   // propagated scalar + self-loop

  float h0 = S * W1[k0]     + b1[k0];     h0 = h0 > 0.0f ? h0 : 0.0f;
  float h1 = S * W1[k0 + 1] + b1[k0 + 1]; h1 = h1 > 0.0f ? h1 : 0.0f;
  v2f a; a.x = h0; a.y = h1;

  v2f b;
  b.x = (m < 4) ? W2[k0 * 4 + m]       : 0.0f;
  b.y = (m < 4) ? W2[(k0 + 1) * 4 + m] : 0.0f;

  v8f c = {};
  c = __builtin_amdgcn_wmma_f32_16x16x4_f32(false, a, false, b, (short)0, c, false, false);

  // Store: lanes 0-3 / 16-19 hold the 4 useful columns (col = m) of rows
  // rowBase..rowBase+7. dis[rowBase+r] is already resident in lane (r + hi*8):
  // fetch via lane permute instead of re-gathering from memory.
  int rowBase = base + (hi ? 8 : 0);
  float dr[8];
#pragma unroll
  for (int r = 0; r < 8; ++r)
    dr[r] = __shfl(dn, r + (hi ? 8 : 0), 32);

  if (base + 16 <= N) {                 // wave-uniform fast path: full tile
    if (m < 4) {
#pragma unroll
      for (int r = 0; r < 8; ++r)
        zz[(rowBase + r) * 4 + m] = c[r] * dr[r];
    }
  } else {                              // tail tile only
    if (m < 4) {
#pragma unroll
      for (int r = 0; r < 8; ++r) {
        int nm = rowBase + r;
        if (nm < N) zz[nm * 4 + m] = c[r] * dr[r];
      }
    }
  }
}

// ---------------- layer-2 vector scatter: agg2[dst][:] += zz[src][:]*dis[dst] ----------------
__global__ void k_edge2(const int* __restrict__ src, const int* __restrict__ dst,
                        const float* __restrict__ zz, const float* __restrict__ dis,
                        float* __restrict__ agg2, int E) {
  int e = blockIdx.x * blockDim.x + threadIdx.x;
  if (e >= E) return;
  __builtin_prefetch(src + e + 8192, 0, 0);
  __builtin_prefetch(dst + e + 8192, 0, 0);
  int s = src[e], d = dst[e];
  float nd = dis[d];
  float4 z = ((const float4*)zz)[s];
  atomicAdd(&agg2[d * 4 + 0], z.x * nd);
  atomicAdd(&agg2[d * 4 + 1], z.y * nd);
  atomicAdd(&agg2[d * 4 + 2], z.z * nd);
  atomicAdd(&agg2[d * 4 + 3], z.w * nd);
}

// ---------------- h2 = relu(agg2 + zz*dis + b2); ts = (h2 @ W3) * dis ----------------
__global__ void k_node2(const float* __restrict__ agg2, const float* __restrict__ zz,
                        const float* __restrict__ dis,
                        const float* __restrict__ b2, const float* __restrict__ W3,
                        float* __restrict__ ts, int N) {
  int i = blockIdx.x * blockDim.x + threadIdx.x;
  if (i >= N) return;
  float dn = dis[i];
  float4 ag = ((const float4*)agg2)[i];
  float4 z  = ((const float4*)zz)[i];
  float h0 = ag.x + z.x * dn + b2[0]; h0 = h0 > 0.0f ? h0 : 0.0f;
  float h1 = ag.y + z.y * dn + b2[1]; h1 = h1 > 0.0f ? h1 : 0.0f;
  float h2 = ag.z + z.z * dn + b2[2]; h2 = h2 > 0.0f ? h2 : 0.0f;
  float h3 = ag.w + z.w * dn + b2[3]; h3 = h3 > 0.0f ? h3 : 0.0f;
  float t = h0 * W3[0] + h1 * W3[1] + h2 * W3[2] + h3 * W3[3];
  ts[i] = t * dn;                      // pre-scale by dis[src]
}

// ---------------- layer-3 scalar scatter: s3[dst] += ts[src]*dis[dst] ----------------
__global__ void k_edge3(const int* __restrict__ src, const int* __restrict__ dst,
                        const float* __restrict__ ts, const float* __restrict__ dis,
                        float* __restrict__ s3, int E) {
  int e = blockIdx.x * blockDim.x + threadIdx.x;
  if (e >= E) return;
  __builtin_prefetch(src + e + 8192, 0, 0);
  __builtin_prefetch(dst + e + 8192, 0, 0);
  int s = src[e], d = dst[e];
  atomicAdd(&s3[d], ts[s] * dis[d]);
}

// ---------------- final node value + pooled segment-sum via LDS binning ----------------
__global__ void k_pool(const float* __restrict__ s3, const float* __restrict__ ts,
                       const float* __restrict__ dis, const float* __restrict__ b3,
                       const int* __restrict__ batch, float* __restrict__ out, int N) {
  __shared__ float pool[NUM_GRAPHS];
  int tid = threadIdx.x;
  for (int j = tid; j < NUM_GRAPHS; j += blockDim.x) pool[j] = 0.0f;
  __syncthreads();

  int i = blockIdx.x * blockDim.x + tid;
  if (i < N) {
    float o = s3[i] + ts[i] * dis[i] + b3[0];
    atomicAdd(&pool[batch[i]], o);     // ds_add_f32; batch sorted -> few hot bins
  }
  __syncthreads();

  for (int j = tid; j < NUM_GRAPHS; j += blockDim.x) {
    float v = pool[j];
    if (v != 0.0f) atomicAdd(&out[j], v);
  }
}

extern "C" void kernel_launch(void* const* d_in, const int* in_sizes, int n_in,
                              void* d_out, int out_size, void* d_ws, size_t ws_size,
                              hipStream_t stream) {
  const float* x     = (const float*)d_in[0];
  const int*   edges = (const int*)  d_in[1];
  const int*   batch = (const int*)  d_in[2];
  const float* W1    = (const float*)d_in[3];
  const float* b1    = (const float*)d_in[4];
  const float* W2    = (const float*)d_in[5];
  const float* b2    = (const float*)d_in[6];
  const float* W3    = (const float*)d_in[7];
  const float* b3    = (const float*)d_in[8];

  const int N = in_sizes[0];
  const int E = in_sizes[1] / 2;
  const int* src = edges;
  const int* dst = edges + E;

  // Workspace layout (16B-aligned regions, Np multiple of 4)
  const int Np = (N + 3) & ~3;
  float* ws   = (float*)d_ws;
  float* degf = ws;            // [Np]   zeroed
  float* s1   = ws + 1 * Np;   // [Np]   zeroed
  float* agg2 = ws + 2 * Np;   // [4Np]  zeroed
  float* s3   = ws + 6 * Np;   // [Np]   zeroed
  float* dis  = ws + 7 * Np;   // [Np]
  float* xs   = ws + 8 * Np;   // [Np]
  float* zz   = ws + 9 * Np;   // [4Np]
  float* ts   = ws + 13 * Np;  // [Np]

  const int B = 256;
  const int gN = (N + B - 1) / B;
  const int gE = (E + B - 1) / B;

  k_zero<<<2048, B, 0, stream>>>(ws, 7 * Np);
  k_zero<<<2,    B, 0, stream>>>((float*)d_out, out_size);

  k_deg  <<<gE, B, 0, stream>>>(dst, degf, E);
  k_norm <<<gN, B, 0, stream>>>(x, degf, dis, xs, N);
  k_edge1<<<gE, B, 0, stream>>>(src, dst, xs, dis, s1, E);

  // 128 nodes per 256-thread block (8 waves x 16 nodes, one WMMA each)
  const int gW = (N + 127) / 128;
  k_l1_wmma<<<gW, B, 0, stream>>>(x, s1, dis, W1, b1, W2, zz, N);

  k_edge2<<<gE, B, 0, stream>>>(src, dst, zz, dis, agg2, E);
  k_node2<<<gN, B, 0, stream>>>(agg2, zz, dis, b2, W3, ts, N);
  k_edge3<<<gE, B, 0, stream>>>(src, dst, ts, dis, s3, E);
  k_pool <<<gN, B, 0, stream>>>(s3, ts, dis, b3, batch, (float*)d_out, N);
}